// MirrorNeRFTcnn_73744588472841
// MI455X (gfx1250) — compile-verified
//
#include <hip/hip_runtime.h>
#include <hip/hip_bf16.h>

// Fused MirrorNeRF forward + density-gradient kernel for gfx1250 (MI455X).
// All MLP GEMMs on the matrix pipe via V_WMMA_F32_16X16X4_F32 (full fp32).
// Hash-grid gathers per-lane (tables L2-resident: 64MB vs 192MB L2).
// Round-2 changes: rolled gather loops (VGPRs < 256, no s_set_vgpr_msb),
// all WMMA operands are single aligned ds_load_b64 (quad-interleaved weight
// staging, even conflict-free activation strides).

typedef __attribute__((ext_vector_type(2))) float v2f;
typedef __attribute__((ext_vector_type(8))) float v8f;

#define NPTS   524288
#define TSIZE  524288
#define TMASK  (TSIZE - 1)
#define WAVES  4
#define PTS_PER_BLOCK (WAVES * 32)

// activation strides (dwords) chosen so lane-addr = idx*stride + k hits
// distinct bank pairs for all 32 lanes on b64 loads
#define LDA_ENC 36
#define LDA_HID 68
#define LDA_GEO 20

__device__ static const int RES_TAB[16] = {16, 22, 30, 42, 58, 80, 111, 153,
                                           212, 294, 406, 561, 776, 1072, 1482, 2048};

__device__ __forceinline__ float sigmoid_f(float v) {
    return 1.0f / (1.0f + __expf(-v));
}

// D(16x16) += A(16xK) * B(KxN-tile); A row-major (lda), B quad-interleaved:
// Bq[(k>>2)*(4N) + n*4 + (k&3)].  One b64 LDS load per operand.
__device__ __forceinline__ v8f wmma_f32(const float* A, int lda,
                                        const float* Bq, int N,
                                        int K, v8f c) {
    const int lane = threadIdx.x & 31;
    const int idx  = lane & 15;
    const int h2   = (lane >> 4) << 1;   // 0 or 2
#pragma unroll
    for (int k0 = 0; k0 < K; k0 += 4) {
        v2f a = *(const v2f*)(A + idx * lda + k0 + h2);
        v2f b = *(const v2f*)(Bq + (k0 >> 2) * (4 * N) + idx * 4 + h2);
        c = __builtin_amdgcn_wmma_f32_16x16x4_f32(false, a, false, b,
                                                  (short)0, c, false, false);
    }
    return c;
}

__global__ __launch_bounds__(128) void mirror_nerf_fused(
    const float* __restrict__ x,     const float* __restrict__ tables,
    const float* __restrict__ ws0,   const float* __restrict__ ws1,
    const float* __restrict__ wc0,   const float* __restrict__ wc1,
    const float* __restrict__ wc2,   const float* __restrict__ wn0,
    const float* __restrict__ wn1,   const float* __restrict__ wm0,
    const float* __restrict__ bm0,   const float* __restrict__ wm1,
    const float* __restrict__ bm1,   float* __restrict__ out)
{
    // ---- weights: quad-interleaved for B-operand b64 loads ----
    __shared__ __attribute__((aligned(16))) float s_ws0[32 * 64];   // K=32,N=64
    __shared__ __attribute__((aligned(16))) float s_ws0t[32 * 68];  // plain [enc][hid], stride 68
    __shared__ __attribute__((aligned(16))) float s_ws1[64 * 16];   // K=64,N=16
    __shared__ __attribute__((aligned(16))) float s_wc0[32 * 64];   // K=32(31+pad),N=64
    __shared__ __attribute__((aligned(16))) float s_wc1[64 * 64];   // K=64,N=64
    __shared__ __attribute__((aligned(16))) float s_wc2[64 * 16];   // K=64,N=16(3+pad)
    __shared__ __attribute__((aligned(16))) float s_wn0[16 * 64];   // K=16(15+pad),N=64
    __shared__ __attribute__((aligned(16))) float s_wn1[64 * 16];   // K=64,N=16(3+pad)
    __shared__ __attribute__((aligned(16))) float s_wm0[16 * 32];   // K=16(15+pad),N=32
    __shared__ __attribute__((aligned(16))) float s_wm1[32 * 16];   // K=32,N=16(1+pad)
    // ---- per-wave activation staging ----
    __shared__ __attribute__((aligned(16))) float s_enc[WAVES][32][LDA_ENC];
    __shared__ __attribute__((aligned(16))) float s_hid[WAVES][32][LDA_HID];
    __shared__ __attribute__((aligned(16))) float s_geo[WAVES][32][LDA_GEO];

    const int tid  = threadIdx.x;
    const int lane = tid & 31;
    const int wave = tid >> 5;
    const int idx  = lane & 15;
    const int half = lane >> 4;
    const int pbase = blockIdx.x * PTS_PER_BLOCK + wave * 32;
    const int p = pbase + lane;

    // ---- stage weights (quad-interleave: dst = (k>>2)*4N + n*4 + (k&3)) ----
    for (int i = tid; i < 32 * 64; i += 128) {            // ws0 (K=32,N=64)
        int k = i >> 6, n = i & 63;
        s_ws0[(k >> 2) * 256 + n * 4 + (k & 3)] = ws0[i];
        s_ws0t[n * 68 + k] = ws0[i];                      // plain transpose-read copy
    }
    for (int i = tid; i < 64 * 16; i += 128) {            // ws1 (K=64,N=16)
        int k = i >> 4, n = i & 15;
        s_ws1[(k >> 2) * 64 + n * 4 + (k & 3)] = ws1[i];
    }
    for (int i = tid; i < 32 * 64; i += 128) {            // wc0 (31x64 pad K)
        int k = i >> 6, n = i & 63;
        s_wc0[(k >> 2) * 256 + n * 4 + (k & 3)] = (k < 31) ? wc0[k * 64 + n] : 0.f;
    }
    for (int i = tid; i < 64 * 64; i += 128) {            // wc1 (K=64,N=64)
        int k = i >> 6, n = i & 63;
        s_wc1[(k >> 2) * 256 + n * 4 + (k & 3)] = wc1[i];
    }
    for (int i = tid; i < 64 * 16; i += 128) {            // wc2 (64x3 pad N)
        int k = i >> 4, n = i & 15;
        s_wc2[(k >> 2) * 64 + n * 4 + (k & 3)] = (n < 3) ? wc2[k * 3 + n] : 0.f;
    }
    for (int i = tid; i < 16 * 64; i += 128) {            // wn0 (15x64 pad K)
        int k = i >> 6, n = i & 63;
        s_wn0[(k >> 2) * 256 + n * 4 + (k & 3)] = (k < 15) ? wn0[k * 64 + n] : 0.f;
    }
    for (int i = tid; i < 64 * 16; i += 128) {            // wn1 (64x3 pad N)
        int k = i >> 4, n = i & 15;
        s_wn1[(k >> 2) * 64 + n * 4 + (k & 3)] = (n < 3) ? wn1[k * 3 + n] : 0.f;
    }
    for (int i = tid; i < 16 * 32; i += 128) {            // wm0 (15x32 pad K)
        int k = i >> 5, n = i & 31;
        s_wm0[(k >> 2) * 128 + n * 4 + (k & 3)] = (k < 15) ? wm0[k * 32 + n] : 0.f;
    }
    for (int i = tid; i < 32 * 16; i += 128) {            // wm1 (32x1 pad N)
        int k = i >> 4, n = i & 15;
        s_wm1[(k >> 2) * 64 + n * 4 + (k & 3)] = (n < 1) ? wm1[k] : 0.f;
    }

    // ---- per-lane point load ----
    const float* xp = x + (size_t)p * 6;
    const float dx = xp[3], dy = xp[4], dz = xp[5];
    const float xn0 = fminf(fmaxf((xp[0] + 1.f) * 0.5f, 0.f), 1.f);
    const float xn1 = fminf(fmaxf((xp[1] + 1.f) * 0.5f, 0.f), 1.f);
    const float xn2 = fminf(fmaxf((xp[2] + 1.f) * 0.5f, 0.f), 1.f);

    // ================= Phase A: hash-grid encode -> s_enc =================
#pragma unroll 1
    for (int l = 0; l < 16; ++l) {
        const int res = RES_TAB[l];
        const float rf = (float)(res - 1);
        const bool dense = (res * res * res) <= TSIZE;
        const float px = xn0 * rf, py = xn1 * rf, pz = xn2 * rf;
        const float fpx = floorf(px), fpy = floorf(py), fpz = floorf(pz);
        const int ix = (int)fpx, iy = (int)fpy, iz = (int)fpz;
        const float fx = px - fpx, fy = py - fpy, fz = pz - fpz;
        const float* tbl = tables + (size_t)l * (TSIZE * 2);
        float f0 = 0.f, f1 = 0.f;
#pragma unroll
        for (int c = 0; c < 8; ++c) {
            const int bx = (c >> 2) & 1, by = (c >> 1) & 1, bz = c & 1;
            const unsigned cx = (unsigned)min(ix + bx, res - 1);
            const unsigned cy = (unsigned)min(iy + by, res - 1);
            const unsigned cz = (unsigned)min(iz + bz, res - 1);
            unsigned id;
            if (dense) id = cx + (unsigned)res * cy + (unsigned)(res * res) * cz;
            else       id = (cx ^ (cy * 2654435761u) ^ (cz * 805459861u)) & TMASK;
            const v2f tv = *(const v2f*)(tbl + (size_t)id * 2u);
            const float wx = bx ? fx : 1.f - fx;
            const float wy = by ? fy : 1.f - fy;
            const float wz = bz ? fz : 1.f - fz;
            const float w = wx * wy * wz;
            f0 = fmaf(w, tv.x, f0);
            f1 = fmaf(w, tv.y, f1);
        }
        v2f fv = { f0, f1 };
        *(v2f*)&s_enc[wave][lane][2 * l] = fv;
    }
    __syncthreads();

    // ================= Phase B: h1 = relu(enc @ ws0) -> s_hid =============
#pragma unroll
    for (int t = 0; t < 2; ++t) {
        const float* A = &s_enc[wave][t * 16][0];
#pragma unroll
        for (int o = 0; o < 4; ++o) {
            v8f acc = {};
            acc = wmma_f32(A, LDA_ENC, &s_ws0[o * 64], 64, 32, acc);
#pragma unroll
            for (int i = 0; i < 8; ++i)
                s_hid[wave][t * 16 + i + 8 * half][o * 16 + idx] = fmaxf(acc[i], 0.f);
        }
    }
    __syncthreads();

    // ====== Phase B2: h = h1 @ ws1 -> sigma, geo_feat (out + s_geo) =======
#pragma unroll
    for (int t = 0; t < 2; ++t) {
        v8f acc = {};
        acc = wmma_f32(&s_hid[wave][t * 16][0], LDA_HID, s_ws1, 16, 64, acc);
#pragma unroll
        for (int i = 0; i < 8; ++i) {
            const int r = t * 16 + i + 8 * half;
            const int pp = pbase + r;
            const float v = acc[i];
            if (idx == 0) {
                out[pp] = v;                                  // sigma
            } else {
                out[NPTS + (size_t)pp * 15 + (idx - 1)] = v;  // geo_feat
                s_geo[wave][r][idx - 1] = v;
            }
        }
    }
    s_geo[wave][lane][15] = 0.f;
    __syncthreads();

    // === Phase C: denc = (mask(h1)*ws1[:,0]) @ ws0^T -> s_enc (overwrite) ==
    {
        const int h2 = half * 2;
#pragma unroll
        for (int t = 0; t < 2; ++t) {
#pragma unroll
            for (int n = 0; n < 2; ++n) {
                v8f acc = {};
#pragma unroll
                for (int k0 = 0; k0 < 64; k0 += 4) {
                    const int ka = k0 + h2;
                    v2f hv = *(const v2f*)(&s_hid[wave][t * 16 + idx][0] + ka);
                    v2f w1 = *(const v2f*)(s_ws1 + (k0 >> 2) * 64 + h2); // ws1[:,0] pair
                    v2f a = { hv.x > 0.f ? w1.x : 0.f,
                              hv.y > 0.f ? w1.y : 0.f };
                    v2f b = *(const v2f*)(&s_ws0t[(n * 16 + idx) * 68 + ka]); // ws0^T
                    acc = __builtin_amdgcn_wmma_f32_16x16x4_f32(
                        false, a, false, b, (short)0, acc, false, false);
                }
#pragma unroll
                for (int i = 0; i < 8; ++i)
                    s_enc[wave][t * 16 + i + 8 * half][n * 16 + idx] = acc[i];
            }
        }
    }
    __syncthreads();

    // ========= Phase D: per-lane gradient re-gather -> normal out =========
    {
        float gx = 0.f, gy = 0.f, gz = 0.f;
#pragma unroll 1
        for (int l = 0; l < 16; ++l) {
            const int res = RES_TAB[l];
            const float rf = (float)(res - 1);
            const bool dense = (res * res * res) <= TSIZE;
            const float px = xn0 * rf, py = xn1 * rf, pz = xn2 * rf;
            const float fpx = floorf(px), fpy = floorf(py), fpz = floorf(pz);
            const int ix = (int)fpx, iy = (int)fpy, iz = (int)fpz;
            const float fx = px - fpx, fy = py - fpy, fz = pz - fpz;
            const v2f dv = *(const v2f*)&s_enc[wave][lane][2 * l];
            const float* tbl = tables + (size_t)l * (TSIZE * 2);
#pragma unroll
            for (int c = 0; c < 8; ++c) {
                const int bx = (c >> 2) & 1, by = (c >> 1) & 1, bz = c & 1;
                const unsigned cx = (unsigned)min(ix + bx, res - 1);
                const unsigned cy = (unsigned)min(iy + by, res - 1);
                const unsigned cz = (unsigned)min(iz + bz, res - 1);
                unsigned id;
                if (dense) id = cx + (unsigned)res * cy + (unsigned)(res * res) * cz;
                else       id = (cx ^ (cy * 2654435761u) ^ (cz * 805459861u)) & TMASK;
                const v2f tv = *(const v2f*)(tbl + (size_t)id * 2u);
                const float cmn = (dv.x * tv.x + dv.y * tv.y) * rf;
                const float wx = bx ? fx : 1.f - fx;
                const float wy = by ? fy : 1.f - fy;
                const float wz = bz ? fz : 1.f - fz;
                const float sx = bx ? 1.f : -1.f;
                const float sy = by ? 1.f : -1.f;
                const float sz = bz ? 1.f : -1.f;
                gx = fmaf(cmn, sx * wy * wz, gx);
                gy = fmaf(cmn, wx * sy * wz, gy);
                gz = fmaf(cmn, wx * wy * sz, gz);
            }
        }
        gx *= 0.5f; gy *= 0.5f; gz *= 0.5f;   // d xn / d xyz = 1/(2*BOUND)
        const float gn = fmaxf(sqrtf(gx * gx + gy * gy + gz * gz), 1e-8f);
        out[16 * (size_t)NPTS + (size_t)p * 3 + 0] = -gx / gn;
        out[16 * (size_t)NPTS + (size_t)p * 3 + 1] = -gy / gn;
        out[16 * (size_t)NPTS + (size_t)p * 3 + 2] = -gz / gn;
    }
    __syncthreads();

    // ======= Phase E1: stage [sh_encode(d), geo_feat] into s_enc ==========
    {
        const float xy = dx * dy, xz = dx * dz, yz = dy * dz;
        const float x2 = dx * dx, y2 = dy * dy, z2 = dz * dz;
        float* row = &s_enc[wave][lane][0];
        row[0]  = 0.28209479177387814f;
        row[1]  = -0.48860251190291987f * dy;
        row[2]  =  0.48860251190291987f * dz;
        row[3]  = -0.48860251190291987f * dx;
        row[4]  =  1.0925484305920792f * xy;
        row[5]  = -1.0925484305920792f * yz;
        row[6]  =  0.94617469575756f * z2 - 0.31539156525252005f;
        row[7]  = -1.0925484305920792f * xz;
        row[8]  =  0.5462742152960396f * (x2 - y2);
        row[9]  =  0.5900435899266435f * dy * (-3.0f * x2 + y2);
        row[10] =  2.890611442640554f * xy * dz;
        row[11] =  0.4570457994644657f * dy * (1.0f - 5.0f * z2);
        row[12] =  0.3731763325901154f * dz * (5.0f * z2 - 3.0f);
        row[13] =  0.4570457994644657f * dx * (1.0f - 5.0f * z2);
        row[14] =  1.445305721320277f * dz * (x2 - y2);
        row[15] =  0.5900435899266435f * dx * (x2 - 3.0f * y2);
#pragma unroll
        for (int j = 0; j < 15; ++j) row[16 + j] = s_geo[wave][lane][j];
        row[31] = 0.f;
    }
    __syncthreads();

    // ============ Phase E2: color layer1 (31->64, relu) -> s_hid ==========
#pragma unroll
    for (int t = 0; t < 2; ++t) {
#pragma unroll
        for (int o = 0; o < 4; ++o) {
            v8f acc = {};
            acc = wmma_f32(&s_enc[wave][t * 16][0], LDA_ENC, &s_wc0[o * 64], 64, 32, acc);
#pragma unroll
            for (int i = 0; i < 8; ++i)
                s_hid[wave][t * 16 + i + 8 * half][o * 16 + idx] = fmaxf(acc[i], 0.f);
        }
    }
    __syncthreads();

    // ===== Phase E3: color layer2 (64->64, relu) in-place reg-buffered ====
#pragma unroll
    for (int t = 0; t < 2; ++t) {
        const float* A = &s_hid[wave][t * 16][0];
        v8f a0 = {}, a1 = {}, a2 = {}, a3 = {};
        a0 = wmma_f32(A, LDA_HID, &s_wc1[0],   64, 64, a0);
        a1 = wmma_f32(A, LDA_HID, &s_wc1[64],  64, 64, a1);
        a2 = wmma_f32(A, LDA_HID, &s_wc1[128], 64, 64, a2);
        a3 = wmma_f32(A, LDA_HID, &s_wc1[192], 64, 64, a3);
#pragma unroll
        for (int i = 0; i < 8; ++i) {
            const int r = t * 16 + i + 8 * half;
            s_hid[wave][r][idx]      = fmaxf(a0[i], 0.f);
            s_hid[wave][r][16 + idx] = fmaxf(a1[i], 0.f);
            s_hid[wave][r][32 + idx] = fmaxf(a2[i], 0.f);
            s_hid[wave][r][48 + idx] = fmaxf(a3[i], 0.f);
        }
    }
    __syncthreads();

    // ====== Phase E4: color layer3 (64->3), sigmoid -> rgb out ============
#pragma unroll
    for (int t = 0; t < 2; ++t) {
        v8f acc = {};
        acc = wmma_f32(&s_hid[wave][t * 16][0], LDA_HID, s_wc2, 16, 64, acc);
#pragma unroll
        for (int i = 0; i < 8; ++i) {
            if (idx < 3) {
                const int pp = pbase + t * 16 + i + 8 * half;
                out[22 * (size_t)NPTS + (size_t)pp * 3 + idx] = sigmoid_f(acc[i]);
            }
        }
    }
    __syncthreads();

    // ======= Phase E5: pred-normal layer1 (15->64, relu) -> s_hid =========
#pragma unroll
    for (int t = 0; t < 2; ++t) {
#pragma unroll
        for (int o = 0; o < 4; ++o) {
            v8f acc = {};
            acc = wmma_f32(&s_geo[wave][t * 16][0], LDA_GEO, &s_wn0[o * 64], 64, 16, acc);
#pragma unroll
            for (int i = 0; i < 8; ++i)
                s_hid[wave][t * 16 + i + 8 * half][o * 16 + idx] = fmaxf(acc[i], 0.f);
        }
    }
    __syncthreads();

    // ===== Phase E6: pred-normal layer2 (64->3), normalize -> out =========
#pragma unroll
    for (int t = 0; t < 2; ++t) {
        v8f acc = {};
        acc = wmma_f32(&s_hid[wave][t * 16][0], LDA_HID, s_wn1, 16, 64, acc);
#pragma unroll
        for (int i = 0; i < 8; ++i)
            if (idx < 3) s_enc[wave][t * 16 + i + 8 * half][idx] = acc[i];
    }
    __syncthreads();
    {
        const float nx = s_enc[wave][lane][0];
        const float ny = s_enc[wave][lane][1];
        const float nz = s_enc[wave][lane][2];
        const float nn = fmaxf(sqrtf(nx * nx + ny * ny + nz * nz), 1e-8f);
        out[19 * (size_t)NPTS + (size_t)p * 3 + 0] = nx / nn;
        out[19 * (size_t)NPTS + (size_t)p * 3 + 1] = ny / nn;
        out[19 * (size_t)NPTS + (size_t)p * 3 + 2] = nz / nn;
    }
    __syncthreads();

    // ===== Phase E7: mirror layer1 (15->32)+bm0, leaky_relu -> s_hid ======
    const float bm1v = bm1[0];
#pragma unroll
    for (int t = 0; t < 2; ++t) {
#pragma unroll
        for (int n = 0; n < 2; ++n) {
            v8f acc = {};
            acc = wmma_f32(&s_geo[wave][t * 16][0], LDA_GEO, &s_wm0[n * 64], 32, 16, acc);
            const float bb = bm0[n * 16 + idx];
#pragma unroll
            for (int i = 0; i < 8; ++i) {
                const float v = acc[i] + bb;
                s_hid[wave][t * 16 + i + 8 * half][n * 16 + idx] =
                    (v > 0.f) ? v : 0.01f * v;
            }
        }
    }
    __syncthreads();

    // ====== Phase E8: mirror layer2 (32->1)+bm1, sigmoid -> out ===========
#pragma unroll
    for (int t = 0; t < 2; ++t) {
        v8f acc = {};
        acc = wmma_f32(&s_hid[wave][t * 16][0], LDA_HID, s_wm1, 16, 32, acc);
        if (idx == 0) {
#pragma unroll
            for (int i = 0; i < 8; ++i)
                out[25 * (size_t)NPTS + pbase + t * 16 + i + 8 * half] =
                    sigmoid_f(acc[i] + bm1v);
        }
    }
}

extern "C" void kernel_launch(void* const* d_in, const int* in_sizes, int n_in,
                              void* d_out, int out_size, void* d_ws, size_t ws_size,
                              hipStream_t stream) {
    const float* x      = (const float*)d_in[0];
    const float* tables = (const float*)d_in[1];
    const float* ws0    = (const float*)d_in[2];
    const float* ws1    = (const float*)d_in[3];
    const float* wc0    = (const float*)d_in[4];
    const float* wc1    = (const float*)d_in[5];
    const float* wc2    = (const float*)d_in[6];
    const float* wn0    = (const float*)d_in[7];
    const float* wn1    = (const float*)d_in[8];
    const float* wm0    = (const float*)d_in[9];
    const float* bm0    = (const float*)d_in[10];
    const float* wm1    = (const float*)d_in[11];
    const float* bm1    = (const float*)d_in[12];
    float* out = (float*)d_out;

    dim3 grid(NPTS / PTS_PER_BLOCK);
    dim3 block(WAVES * 32);
    hipLaunchKernelGGL(mirror_nerf_fused, grid, block, 0, stream,
                       x, tables, ws0, ws1, wc0, wc1, wc2,
                       wn0, wn1, wm0, bm0, wm1, bm1, out);
}